// GCN_12489764897129
// MI455X (gfx1250) — compile-verified
//
#include <hip/hip_runtime.h>

#define N_NODES 50000
#define N_EDGES 800000
#define IN_FT   256
#define OUT_FT  128

typedef __attribute__((ext_vector_type(2))) float v2f;
typedef __attribute__((ext_vector_type(8))) float v8f;

#define M_TILES (N_NODES / 16)   // 3125
#define N_TILES (OUT_FT / 16)    // 8

// ---------------------------------------------------------------------------
// Kernel 1: zero the f32 accumulator (d_out) before atomic scatter.
// ---------------------------------------------------------------------------
__global__ void gcn_zero(float* __restrict__ out, int n) {
    int i = blockIdx.x * blockDim.x + threadIdx.x;
    if (i < n) out[i] = 0.0f;
}

// ---------------------------------------------------------------------------
// Kernel 2: seq_fts = seq @ W^T via V_WMMA_F32_16X16X4_F32.
//
// One wave32 computes a full 16x128 output row-block (8 N-tiles), holding
// 8 x v8f accumulators (64 VGPRs). Per K-pair (K+=8): 2 A-fragment loads
// (seq read once per M-tile -> 51.2 MB total) and 16 B-fragment loads from
// W (128 KB, L2-hot) at immediate offsets. Loads are batched into distinct
// registers BEFORE the WMMA bursts so the scheduler can clause them and
// overlap memory with the matrix pipe instead of wait-0 serializing.
//
// Fragment layouts (ISA 7.12.2):
//  A (16x4 f32, 2 VGPRs): lanes 0-15 row M=lane, K={0,1}; lanes 16-31 K={2,3}
//  B (4x16 f32, 2 VGPRs): lane col N=lane&15; VGPR0 row K0/K2, VGPR1 K1/K3
//  C/D (16x16 f32, 8 VGPRs): VGPR v -> row v (lanes<16) / v+8 (lanes>=16)
// ---------------------------------------------------------------------------
__global__ void __launch_bounds__(128, 1)
gcn_gemm_wmma(const float* __restrict__ seq,
              const float* __restrict__ W,
              float* __restrict__ fts) {
    const int lane = threadIdx.x & 31;
    const int wave = threadIdx.x >> 5;
    const int mt   = blockIdx.x * (blockDim.x >> 5) + wave;
    if (mt >= M_TILES) return;                // wave-uniform guard

    const int half = lane >> 4;               // 0: lanes 0-15, 1: lanes 16-31
    const int khi  = half * 2;                // K sub-offset {0,2}
    const int mrow = mt * 16 + (lane & 15);   // A row supplied by this lane

    const float* __restrict__ arow  = seq + (size_t)mrow * IN_FT;
    // B base for this lane's column (local col = lane&15); N-tile selected by
    // an immediate byte offset nt*16*IN_FT*4 = nt*16384 (< 24-bit IOFFSET).
    const float* __restrict__ bbase = W + (size_t)(lane & 15) * IN_FT;

    v8f c[N_TILES];
    #pragma unroll
    for (int t = 0; t < N_TILES; ++t) c[t] = (v8f){};

    #pragma unroll 2
    for (int k = 0; k < IN_FT; k += 8) {
        // ---- batch ALL loads for two K-steps into independent registers ----
        v2f a0, a1;
        a0 = *(const v2f*)(arow + k + khi);
        a1 = *(const v2f*)(arow + k + 4 + khi);

        v2f b0[N_TILES], b1[N_TILES];
        #pragma unroll
        for (int nt = 0; nt < N_TILES; ++nt)
            b0[nt] = *(const v2f*)(bbase + nt * 16 * IN_FT + k + khi);
        #pragma unroll
        for (int nt = 0; nt < N_TILES; ++nt)
            b1[nt] = *(const v2f*)(bbase + nt * 16 * IN_FT + k + 4 + khi);

        // ---- 16 WMMAs on fully-loaded fragments ----
        #pragma unroll
        for (int nt = 0; nt < N_TILES; ++nt)
            c[nt] = __builtin_amdgcn_wmma_f32_16x16x4_f32(
                        false, a0, false, b0[nt], (short)0, c[nt], false, false);
        #pragma unroll
        for (int nt = 0; nt < N_TILES; ++nt)
            c[nt] = __builtin_amdgcn_wmma_f32_16x16x4_f32(
                        false, a1, false, b1[nt], (short)0, c[nt], false, false);
    }

    // Store D tiles: VGPR v holds row (mt*16 + v + 8*half), col nt*16+(lane&15)
    const int mbase = mt * 16 + half * 8;
    float* __restrict__ orow = fts + (size_t)mbase * OUT_FT + (lane & 15);
    #pragma unroll
    for (int v = 0; v < 8; ++v) {
        #pragma unroll
        for (int nt = 0; nt < N_TILES; ++nt) {
            orow[(size_t)v * OUT_FT + nt * 16] = c[nt][v];
        }
    }
}

// ---------------------------------------------------------------------------
// Kernel 3: out[dst] += val * seq_fts[src]. One wave32 per edge; each lane
// gathers a float4 (lane*4 .. lane*4+3 of the 128 features) and issues 4
// global f32 atomic adds (accumulator is 25.6 MB, L2-resident).
// ---------------------------------------------------------------------------
__global__ void gcn_scatter(const float* __restrict__ fts,
                            const int*   __restrict__ esrc,
                            const int*   __restrict__ edst,
                            const float* __restrict__ eval,
                            float* __restrict__ out) {
    const int gwave = (blockIdx.x * blockDim.x + threadIdx.x) >> 5;
    const int lane  = threadIdx.x & 31;
    if (gwave >= N_EDGES) return;

    const int   s = esrc[gwave];
    const int   d = edst[gwave];
    const float v = eval[gwave];

    const float4 m = *((const float4*)(fts + (size_t)s * OUT_FT) + lane);
    float* o = out + (size_t)d * OUT_FT + lane * 4;
    atomicAdd(o + 0, v * m.x);
    atomicAdd(o + 1, v * m.y);
    atomicAdd(o + 2, v * m.z);
    atomicAdd(o + 3, v * m.w);
}

// ---------------------------------------------------------------------------
// Kernel 4: out = PReLU(out + bias)
// ---------------------------------------------------------------------------
__global__ void gcn_epilogue(float* __restrict__ out,
                             const float* __restrict__ bias,
                             const float* __restrict__ prelu_a,
                             int n) {
    int i = blockIdx.x * blockDim.x + threadIdx.x;
    if (i < n) {
        float x = out[i] + bias[i & (OUT_FT - 1)];
        float a = prelu_a[0];
        out[i] = (x >= 0.0f) ? x : a * x;
    }
}

// ---------------------------------------------------------------------------
// Input order (setup_inputs dict): seq, edge_src, edge_dst, edge_val, W,
// bias, prelu_a. Output: [N_NODES, OUT_FT] f32. d_ws holds seq_fts (25.6 MB).
// ---------------------------------------------------------------------------
extern "C" void kernel_launch(void* const* d_in, const int* in_sizes, int n_in,
                              void* d_out, int out_size, void* d_ws, size_t ws_size,
                              hipStream_t stream) {
    const float* seq     = (const float*)d_in[0];
    const int*   esrc    = (const int*)  d_in[1];
    const int*   edst    = (const int*)  d_in[2];
    const float* eval    = (const float*)d_in[3];
    const float* W       = (const float*)d_in[4];
    const float* bias    = (const float*)d_in[5];
    const float* prelu_a = (const float*)d_in[6];

    float* out = (float*)d_out;
    float* fts = (float*)d_ws;            // [N_NODES, OUT_FT] scratch

    const int n_out_elems = N_NODES * OUT_FT;   // 6.4M

    // 1) zero accumulator
    gcn_zero<<<(n_out_elems + 255) / 256, 256, 0, stream>>>(out, n_out_elems);

    // 2) WMMA GEMM: 3125 M-blocks, one wave each, 4 waves per 128-thread block
    {
        const int blocks = (M_TILES + 3) / 4;              // 782
        gcn_gemm_wmma<<<blocks, 128, 0, stream>>>(seq, W, fts);
    }

    // 3) edge scatter: 800000 waves, 8 waves per 256-thread block
    {
        const int blocks = (N_EDGES + 7) / 8;              // 100000
        gcn_scatter<<<blocks, 256, 0, stream>>>(fts, esrc, edst, eval, out);
    }

    // 4) bias + PReLU
    gcn_epilogue<<<(n_out_elems + 255) / 256, 256, 0, stream>>>(out, bias, prelu_a, n_out_elems);
}